// LongTermROUNDModel_9861244911785
// MI455X (gfx1250) — compile-verified
//
#include <hip/hip_runtime.h>
#include <math.h>

// Problem dims (fixed by the reference)
#define B_  128
#define S_  1024
#define D_  64
#define H_  128
#define O_  512
#define KF_ 896          // 7*H feature dimension

#define TWO_PI      6.28318530717958647692f
#define INV_TWO_PI  0.15915494309189533577f

typedef float v2f __attribute__((ext_vector_type(2)));
typedef float v8f __attribute__((ext_vector_type(8)));

// ---------------------------------------------------------------------------
// Kernel 0: rep[s] = all(x[:,s,:] == x[:,s-1,:])  (global scalar per step)
// ---------------------------------------------------------------------------
__global__ __launch_bounds__(256)
void rep_kernel(const float* __restrict__ x, int* __restrict__ rep) {
    const int s = blockIdx.x;
    __shared__ int sflag;
    if (threadIdx.x == 0) sflag = 1;
    __syncthreads();
    if (s > 0) {
        bool ok = true;
        for (int i = threadIdx.x; i < B_ * D_; i += 256) {
            const int b = i >> 6, d = i & 63;
            const float cur = x[((size_t)b * S_ + s)     * D_ + d];
            const float prv = x[((size_t)b * S_ + (s-1)) * D_ + d];
            ok = ok && (cur == prv);
        }
        if (!ok) atomicAnd(&sflag, 0);
    }
    __syncthreads();
    if (threadIdx.x == 0) rep[s] = (s == 0) ? 0 : sflag;
}

// ---------------------------------------------------------------------------
// Kernel T: transpose W[n][k] -> paired-K layout  WT[(k>>1)*N + n][2] so a
// B-fragment (lane=n, values k,k+1) is ONE fully-coalesced b64 load.
// One-time cost: Wr 1.8 MB, We 32 KB.
// ---------------------------------------------------------------------------
__global__ __launch_bounds__(256)
void transpose_pairk(const float* __restrict__ src, float* __restrict__ dst,
                     int nrows, int ncols) {           // src is [nrows][ncols]
    const int e = blockIdx.x * 256 + threadIdx.x;
    if (e >= nrows * ncols) return;
    const int n = e / ncols;
    const int k = e - n * ncols;
    dst[((size_t)(k >> 1) * nrows + n) * 2 + (k & 1)] = src[e];
}

// ---------------------------------------------------------------------------
// Kernel 1: theta[row,h] = wrap(x[row,:]·We[h,:] + be[h]),  row = b*S+s
// One block = one M-tile (16 rows). 8 waves, wave w owns h-tile w (H=8*16).
// WMMA f32 16x16x4, K = 64 -> 16 k-steps. B from paired-K WeT (coalesced b64).
// ---------------------------------------------------------------------------
__global__ __launch_bounds__(256)
void encoder_kernel(const float* __restrict__ x, const float* __restrict__ WeT,
                    const float* __restrict__ be, float* __restrict__ theta) {
    __shared__ float xs[16][68];               // stride 68: 16 rows -> 16 distinct banks
    const int tid  = threadIdx.x;
    const int wave = tid >> 5;
    const int lane = tid & 31;
    const size_t row0 = (size_t)blockIdx.x * 16;

    // coalesced stage of the 16x64 A tile
    for (int e = tid; e < 16 * 64; e += 256) {
        const int r = e >> 6, c = e & 63;
        xs[r][c] = x[(row0 + r) * D_ + c];
    }
    __syncthreads();

    const int m     = lane & 15;               // A row / output column lane
    const int khalf = (lane >> 4) << 1;        // 0 for lanes 0-15, 2 for 16-31
    const int n     = wave * 16 + m;           // output h index for this lane
    const v2f* Wp   = (const v2f*)WeT;         // [D/2][H] pairs

    v8f acc = {0.f, 0.f, 0.f, 0.f, 0.f, 0.f, 0.f, 0.f};
    for (int k0 = 0; k0 < D_; k0 += 4) {
        const int kk = k0 + khalf;
        v2f a;
        a.x = xs[m][kk];
        a.y = xs[m][kk + 1];
        const v2f b = Wp[(size_t)(kk >> 1) * H_ + n];   // {We[n][kk], We[n][kk+1]}
        acc = __builtin_amdgcn_wmma_f32_16x16x4_f32(
                  false, a, false, b, (short)0, acc, false, false);
    }

    const float bias = be[n];
    const int   mhi  = (lane >> 4) << 3;       // +8 rows for upper half-wave
    #pragma unroll
    for (int v = 0; v < 8; ++v) {
        const float pt = acc[v] + bias;
        // atan2(sin(pt),cos(pt)) == wrap to (-pi,pi]
        const float th = pt - TWO_PI * rintf(pt * INV_TWO_PI);
        theta[(row0 + v + mhi) * H_ + n] = th;
    }
}

// ---------------------------------------------------------------------------
// Kernel 2: sequential phase recurrence. One thread per (b,h) chain.
// ---------------------------------------------------------------------------
__global__ __launch_bounds__(256)
void scan_kernel(const float* __restrict__ theta, const float* __restrict__ omega,
                 const int* __restrict__ rep,
                 float* __restrict__ ph_hist, float* __restrict__ wb_hist) {
    const int t = blockIdx.x * 256 + threadIdx.x;   // 0 .. B*H-1
    const int h = t & (H_ - 1);
    const int b = t >> 7;
    const float om = omega[h];
    float ph = 0.f, wb = 0.f;
    size_t idx = (size_t)b * S_ * H_ + h;
    for (int s = 0; s < S_; ++s, idx += H_) {
        const float th = theta[idx];
        wb += 0.015625f;                             // WOBBLE_STEP
        // ph += omega + COUPLING*sin(th-ph) + AMP*sin(wb), COUPLING=-1, AMP=0.25
        ph = ph + om - __sinf(th - ph) + 0.25f * __sinf(wb);
        if (rep[s]) wb += 0.25f * __sinf(th - wb);   // WOBBLE_COUPLING
        ph_hist[idx] = ph;
        wb_hist[idx] = wb;
    }
}

// ---------------------------------------------------------------------------
// Kernel 3: logits = feats·Wrᵀ + br, feats built on the fly in LDS.
// One block = one M-tile (16 rows) x all of N=512. 8 waves x 4 N-tiles each.
// WMMA f32 16x16x4, K = 896 -> 224 k-steps; one A-frag feeds 4 WMMAs;
// B-fragments are coalesced b64 loads from paired-K WrT.
// ---------------------------------------------------------------------------
__global__ __launch_bounds__(256)
void readout_kernel(const float* __restrict__ ph_hist, const float* __restrict__ wb_hist,
                    const float* __restrict__ WrT, const float* __restrict__ br,
                    float* __restrict__ logits) {
    __shared__ float fs[16][900];              // 57.6 KB; stride 900 -> bank-conflict free
    const int tid  = threadIdx.x;
    const int wave = tid >> 5;
    const int lane = tid & 31;
    const size_t row0 = (size_t)blockIdx.x * 16;

    // Build the 16 x 896 feature tile (coalesced over h within each block of 128)
    for (int e = tid; e < 16 * KF_; e += 256) {
        const int r   = e / KF_;
        const int f   = e - r * KF_;
        const int blk = f >> 7;                // 0..6
        const int h   = f & (H_ - 1);
        const size_t idx = (row0 + r) * H_ + h;
        float val;
        if (blk < 4) {
            const float p = ph_hist[idx];
            val = (blk == 0) ? __cosf(p)
                : (blk == 1) ? __sinf(p)
                : (blk == 2) ? __cosf(0.5f * p)
                             : __sinf(0.5f * p);
        } else if (blk < 6) {
            const float w = wb_hist[idx];
            val = (blk == 4) ? __cosf(w) : __sinf(w);
        } else {
            val = ph_hist[idx];
        }
        fs[r][f] = val;
    }
    __syncthreads();

    const int m     = lane & 15;
    const int khalf = (lane >> 4) << 1;
    const int nbase = wave * 64 + m;           // wave owns N-tiles 4w..4w+3
    const v2f* Wp   = (const v2f*)WrT;         // [KF/2][O] pairs

    v8f acc0 = {0.f,0.f,0.f,0.f,0.f,0.f,0.f,0.f};
    v8f acc1 = acc0, acc2 = acc0, acc3 = acc0;

    for (int k0 = 0; k0 < KF_; k0 += 4) {
        const int kk = k0 + khalf;
        v2f a;
        a.x = fs[m][kk];
        a.y = fs[m][kk + 1];
        const size_t bidx = (size_t)(kk >> 1) * O_;   // pair-row base
        const v2f b0 = Wp[bidx + nbase];              // lanes -> consecutive 8B: coalesced
        const v2f b1 = Wp[bidx + nbase + 16];
        const v2f b2 = Wp[bidx + nbase + 32];
        const v2f b3 = Wp[bidx + nbase + 48];
        acc0 = __builtin_amdgcn_wmma_f32_16x16x4_f32(false, a, false, b0, (short)0, acc0, false, false);
        acc1 = __builtin_amdgcn_wmma_f32_16x16x4_f32(false, a, false, b1, (short)0, acc1, false, false);
        acc2 = __builtin_amdgcn_wmma_f32_16x16x4_f32(false, a, false, b2, (short)0, acc2, false, false);
        acc3 = __builtin_amdgcn_wmma_f32_16x16x4_f32(false, a, false, b3, (short)0, acc3, false, false);
    }

    const int mhi = (lane >> 4) << 3;
    #pragma unroll
    for (int j = 0; j < 4; ++j) {
        const int n  = nbase + j * 16;
        const float bb = br[n];
        const v8f acc = (j == 0) ? acc0 : (j == 1) ? acc1 : (j == 2) ? acc2 : acc3;
        #pragma unroll
        for (int v = 0; v < 8; ++v) {
            logits[(row0 + v + mhi) * O_ + n] = acc[v] + bb;
        }
    }
}

// ---------------------------------------------------------------------------
extern "C" void kernel_launch(void* const* d_in, const int* in_sizes, int n_in,
                              void* d_out, int out_size, void* d_ws, size_t ws_size,
                              hipStream_t stream) {
    const float* x  = (const float*)d_in[0];   // [B,S,D]
    const float* We = (const float*)d_in[1];   // [H,D]
    const float* be = (const float*)d_in[2];   // [H]
    const float* om = (const float*)d_in[3];   // [H]
    const float* Wr = (const float*)d_in[4];   // [O,7H]
    const float* br = (const float*)d_in[5];   // [O]

    float* out     = (float*)d_out;
    float* logits  = out;                                  // [B,S,O]
    float* ph_hist = out + (size_t)B_ * S_ * O_;           // [B,S,H]
    float* wb_hist = ph_hist + (size_t)B_ * S_ * H_;       // [B,S,H]

    // theta scratch lives in the (not-yet-written) logits region: 67 MB <= 268 MB,
    // and is dead before readout_kernel overwrites logits.
    float* theta = logits;

    // workspace: [0,4KB) rep flags | WrT (KF*O floats) | WeT (D*H floats)
    int*   rep = (int*)d_ws;
    float* WrT = (float*)((char*)d_ws + 4096);
    float* WeT = WrT + (size_t)KF_ * O_;

    rep_kernel     <<<S_,                    256, 0, stream>>>(x, rep);
    transpose_pairk<<<(O_*KF_ + 255) / 256,  256, 0, stream>>>(Wr, WrT, O_, KF_);
    transpose_pairk<<<(H_*D_  + 255) / 256,  256, 0, stream>>>(We, WeT, H_, D_);
    encoder_kernel <<<(B_*S_) / 16,          256, 0, stream>>>(x, WeT, be, theta);
    scan_kernel    <<<(B_*H_) / 256,         256, 0, stream>>>(theta, om, rep, ph_hist, wb_hist);
    readout_kernel <<<(B_*S_) / 16,          256, 0, stream>>>(ph_hist, wb_hist, WrT, br, logits);
}